// Router_31705448579443
// MI455X (gfx1250) — compile-verified
//
#include <hip/hip_runtime.h>
#include <hip/hip_bf16.h>
#include <stdint.h>

// ---------------------------------------------------------------------------
// MoE router: logits = x @ W^T  (x:[16384,2048] f32, W:[64,2048] f32)
// then top-2 + softmax.  HBM-bound problem (~134 MB of x at 23.3 TB/s).
// Strategy: f32 WMMA (V_WMMA_F32_16X16X4_F32), one wave per 16-token tile
// covering all 64 experts (4 C tiles).  W double-buffered in LDS via the
// CDNA5 async global->LDS path (no staging VGPRs, tracked by ASYNCcnt).
// ---------------------------------------------------------------------------

typedef __attribute__((ext_vector_type(2))) float v2f;
typedef __attribute__((ext_vector_type(4))) float v4f;
typedef __attribute__((ext_vector_type(8))) float v8f;

#define TOKENS   16384
#define DMODEL   2048
#define NEXPERT  64
#define KC       128           // K chunk staged in LDS
#define LDS_ROW  (KC + 4)      // +4 floats pad: bank stride 132%64=4 -> conflict-free
#define NCHUNK   (DMODEL / KC) // 16

// Async copy 16 B from global to LDS (GV mode), tracked by ASYNCcnt.
// VDST = VGPR holding LDS byte address (low 32 bits of the generic pointer,
// per the LDS aperture mapping), VADDR = 64-bit global address pair.
__device__ __forceinline__ void async_cp_b128(const float* lds_dst, const float* gsrc) {
    const uint32_t lds_off = (uint32_t)(uintptr_t)lds_dst;
    const uint64_t gaddr   = (uint64_t)(uintptr_t)gsrc;
    asm volatile("global_load_async_to_lds_b128 %0, %1, off"
                 :: "v"(lds_off), "v"(gaddr)
                 : "memory");
}

__device__ __forceinline__ void wait_async0() {
    asm volatile("s_wait_asynccnt 0x0" ::: "memory");
}

__global__ __launch_bounds__(256)
void router_wmma_f32(const float* __restrict__ x,
                     const float* __restrict__ W,
                     int*   __restrict__ topi,
                     float* __restrict__ wout) {
    // Double-buffered W chunk: 2 * 64 * 132 * 4B = 67,584 B of LDS (<< 320 KB)
    __shared__ float lds[2][NEXPERT * LDS_ROW];

    const int tid  = threadIdx.x;
    const int lane = tid & 31;
    const int wave = tid >> 5;
    const int mbase = blockIdx.x * 128 + wave * 16;   // 8 waves * 16 tokens

    // staging assignment: thread t copies float4 (e = i*8 + t/32, k = (t&31)*4)
    // global side: consecutive lanes -> consecutive float4 of one expert row
    // LDS side: row stride 528 B (16B-aligned), conflict-free pattern
    const int se_sub = tid >> 5;
    const int sk     = (tid & 31) * 4;

    // A fragment addressing (f32 A 16x4): lanes 0-15 hold K={0,1}, 16-31 K={2,3}
    const float* xrow = x + (size_t)(mbase + (lane & 15)) * DMODEL + ((lane >> 4) * 2);

    // B fragment addressing (f32 B 4x16): lane n = lane&15 (+16*nt), k pair
    const int bofs = (lane & 15) * LDS_ROW + ((lane >> 4) * 2);

    v8f acc0 = {}, acc1 = {}, acc2 = {}, acc3 = {};

    // Prologue: async-stage chunk 0 into lds[0]
#pragma unroll
    for (int i = 0; i < 8; ++i) {
        const int e = i * 8 + se_sub;
        async_cp_b128(&lds[0][e * LDS_ROW + sk], W + (size_t)e * DMODEL + sk);
    }
    wait_async0();
    __syncthreads();

    int buf = 0;
    for (int c = 0; c < NCHUNK; ++c) {
        // After the barrier every wave is done reading lds[buf^1]: safe to
        // overlap the async refill of it with this chunk's compute.
        if (c + 1 < NCHUNK) {
            const float* Wc = W + (size_t)(c + 1) * KC;
#pragma unroll
            for (int i = 0; i < 8; ++i) {
                const int e = i * 8 + se_sub;
                async_cp_b128(&lds[buf ^ 1][e * LDS_ROW + sk],
                              Wc + (size_t)e * DMODEL + sk);
            }
        }

        const float* xc = xrow + c * KC;
        const float* bb = &lds[buf][bofs];

#pragma unroll 4
        for (int ks = 0; ks < KC / 4; ++ks) {
            // A: 16 tokens x 4 K (x read exactly once device-wide)
            v2f a  = *(const v2f*)(xc + ks * 4);
            // B: 4 N-tiles of 16 experts each, from LDS (ds_load_b64, no conflicts)
            v2f b0 = *(const v2f*)(bb + 0 * 16 * LDS_ROW + ks * 4);
            v2f b1 = *(const v2f*)(bb + 1 * 16 * LDS_ROW + ks * 4);
            v2f b2 = *(const v2f*)(bb + 2 * 16 * LDS_ROW + ks * 4);
            v2f b3 = *(const v2f*)(bb + 3 * 16 * LDS_ROW + ks * 4);
            acc0 = __builtin_amdgcn_wmma_f32_16x16x4_f32(false, a, false, b0, (short)0, acc0, false, false);
            acc1 = __builtin_amdgcn_wmma_f32_16x16x4_f32(false, a, false, b1, (short)0, acc1, false, false);
            acc2 = __builtin_amdgcn_wmma_f32_16x16x4_f32(false, a, false, b2, (short)0, acc2, false, false);
            acc3 = __builtin_amdgcn_wmma_f32_16x16x4_f32(false, a, false, b3, (short)0, acc3, false, false);
        }

        if (c + 1 < NCHUNK) wait_async0();   // refill of buf^1 complete
        __syncthreads();
        buf ^= 1;
    }

    // ---------------- Epilogue: per-token top-2 + softmax --------------------
    // C layout: lane L, reg v -> token m = v + 8*(L>>4), expert n = (L&15) + 16*nt.
    // For fixed v: each 16-lane half holds one token's 64 logits (4 per lane).
    const int nlo = lane & 15;
    const float NEG = -3.402823466e38f;

#pragma unroll
    for (int v = 0; v < 8; ++v) {
        float cv[4] = { acc0[v], acc1[v], acc2[v], acc3[v] };
        // local top-2 of this lane's 4 experts (ascending expert ids -> strict >
        // keeps the lower index on ties, matching jax.lax.top_k)
        float v1 = cv[0], v2 = NEG;
        int   e1 = nlo,   e2 = 0;
#pragma unroll
        for (int j = 1; j < 4; ++j) {
            const float val = cv[j];
            const int   eid = nlo + 16 * j;
            if (val > v1)      { v2 = v1; e2 = e1; v1 = val; e1 = eid; }
            else if (val > v2) { v2 = val; e2 = eid; }
        }
        // cross-lane top-2 merge within each 16-lane half (xor masks stay in-half)
#pragma unroll
        for (int mask = 1; mask <= 8; mask <<= 1) {
            const float ov1 = __shfl_xor(v1, mask, 32);
            const int   oe1 = __shfl_xor(e1, mask, 32);
            const float ov2 = __shfl_xor(v2, mask, 32);
            const int   oe2 = __shfl_xor(e2, mask, 32);
            const bool firstA = (v1 > ov1) || (v1 == ov1 && e1 < oe1);
            const float n1 = firstA ? v1 : ov1;
            const int  ne1 = firstA ? e1 : oe1;
            const float ca = firstA ? ov1 : v1;   // loser's best
            const int  cea = firstA ? oe1 : e1;
            const float cb = firstA ? v2 : ov2;   // winner's second
            const int  ceb = firstA ? e2 : oe2;
            const bool secA = (ca > cb) || (ca == cb && cea < ceb);
            v2 = secA ? ca : cb;
            e2 = secA ? cea : ceb;
            v1 = n1; e1 = ne1;
        }
        if (nlo == 0) {
            const int t = mbase + v + ((lane >> 4) << 3);
            const float ed = __expf(v2 - v1);        // <= 1, numerically safe
            const float inv = 1.0f / (1.0f + ed);
            topi[2 * t]     = e1;
            topi[2 * t + 1] = e2;
            wout[2 * t]     = inv;
            wout[2 * t + 1] = ed * inv;
        }
    }
}

extern "C" void kernel_launch(void* const* d_in, const int* in_sizes, int n_in,
                              void* d_out, int out_size, void* d_ws, size_t ws_size,
                              hipStream_t stream) {
    const float* x = (const float*)d_in[0];   // [16384, 2048] f32
    const float* W = (const float*)d_in[1];   // [64, 2048] f32
    // d_out = topi (int32, 16384*2) then weights (f32, 16384*2), concatenated flat
    int*   topi = (int*)d_out;
    float* wout = (float*)d_out + 2 * TOKENS;
    router_wmma_f32<<<TOKENS / 128, 256, 0, stream>>>(x, W, topi, wout);
}